// MyFilter_66735201845567
// MI455X (gfx1250) — compile-verified
//
#include <hip/hip_runtime.h>

// ---------------------------------------------------------------------------
// Bicubic crop-resize (256 -> 512, scale exactly 2) as banded WMMA matmuls.
//
// Roofline: ~250 MB traffic @ 23.3 TB/s => ~11us floor; compute is ~2.4 GFLOP
// done by v_wmma_f32_16x16x4_f32 (65536 waves * 18 wmma) => negligible.
// Out = V * (P * H^T): V,H are banded 4-tap Catmull-Rom (a=-0.5) matrices;
// per 16x16 output tile both factors are 16x12 = three K=4 f32 WMMAs, full
// fp32 precision, jax.image.resize edge semantics (truncate + renormalize).
//
// v2: weight fragments come from a 24KB precomputed __device__ table (one
// tiny init kernel per launch) instead of per-lane select chains; LDS region
// padded to 16 rows so all fragment reads are unconditional ds_load_b64.
// ---------------------------------------------------------------------------

typedef __attribute__((ext_vector_type(2))) float v2f;
typedef __attribute__((ext_vector_type(8))) float v8f;

#define IMG_H 512
#define IMG_W 512
#define NCH   3
#define PATCH 256
#define NB    64          // batch
#define TPI   32          // 16-wide tiles per image dimension (512/16)

// Weight table: [tile 0..31][v 0..15][k 0..11] — identical for both axes.
__device__ float g_wtab[TPI * 16 * 12];

// wave32 cross-lane gather: dst[lane] = src[srcLane]
__device__ __forceinline__ float laneShuffle(float v, int srcLane) {
  return __builtin_bit_cast(
      float, __builtin_amdgcn_ds_bpermute(srcLane << 2,
                                          __builtin_bit_cast(int, v)));
}

// Weights of output row/col (tile-local v, tile index) over the 12-wide input
// band starting at patch coordinate tile*8-2. Phase f=0.25 (odd) / 0.75
// (even); taps outside [0,PATCH) dropped, row renormalized (jax semantics).
__global__ void build_weight_table() {
  const int idx = blockIdx.x * blockDim.x + threadIdx.x;  // 0..511
  if (idx >= TPI * 16) return;
  const int tile = idx >> 4;
  const int v    = idx & 15;

  const int odd = v & 1;
  const int k0  = (v + odd) >> 1;
  const float t0 = odd ? -0.0703125f : -0.0234375f;
  const float t1 = odd ?  0.8671875f :  0.2265625f;
  const float t2 = odd ?  0.2265625f :  0.8671875f;
  const float t3 = odd ? -0.0234375f : -0.0703125f;
  const int gbase = tile * 8 - 2;

  float w[12];
  float sum = 0.f;
#pragma unroll
  for (int k = 0; k < 12; ++k) {
    const int d  = k - k0;
    const int gp = gbase + k;
    float wk = (d == 0) ? t0 : (d == 1) ? t1 : (d == 2) ? t2 : (d == 3) ? t3 : 0.f;
    if (gp < 0 || gp >= PATCH) wk = 0.f;
    w[k] = wk;
    sum += wk;
  }
  const float inv = 1.0f / sum;
  float* dst = g_wtab + idx * 12;
#pragma unroll
  for (int k = 0; k < 12; ++k) dst[k] = w[k] * inv;
}

__global__ __launch_bounds__(256) void crop_resize_bicubic_wmma(
    const float* __restrict__ img,   // [64,512,512,3] f32
    const int*   __restrict__ coord, // [64,3] = {256, i, j}
    float*       __restrict__ out)   // [64,512,512,3] f32
{
  // per-wave channel-planar band: [ch][row 0..15][col 0..11]; rows 12..15 are
  // never staged -> garbage, but only feed T rows 12..15 which the vertical
  // pass (K=12) never consumes.
  __shared__ float lds[8 * NCH * 16 * 12];

  const int lane = threadIdx.x & 31;
  const int wave = threadIdx.x >> 5;
  const int tileId = blockIdx.x * 8 + wave;           // 0..65535, exact
  const int b  = tileId >> 10;                        // image
  const int ty = (tileId >> 5) & 31;                  // tile row
  const int tx = tileId & 31;                         // tile col

  const int ci = coord[b * 3 + 1];                    // crop row offset
  const int cj = coord[b * 3 + 2];                    // crop col offset

  // ---- stage the 12x12x3 patch band into LDS (clamped addresses; taps
  //      outside the patch have zero weight, so clamped data is inert) ------
  float* L = lds + wave * (NCH * 16 * 12);
  const int r0 = ci + ty * 8 - 2;
  const int c0 = cj + tx * 8 - 2;
#pragma unroll
  for (int i = 0; i < 14; ++i) {                      // 448 = 14*32 >= 432
    const int e   = i * 32 + lane;
    const int ch  = e / 144;                          // 0..3 (3 harmless)
    const int rem = e - ch * 144;
    const int row = rem / 12;
    const int col = rem - row * 12;
    const int ir = min(max(r0 + row, 0), IMG_H - 1);
    const int ic = min(max(c0 + col, 0), IMG_W - 1);
    const int chc = min(ch, NCH - 1);
    L[chc * 192 + row * 12 + col] =
        img[((b * IMG_H + ir) * IMG_W + ic) * NCH + chc];
  }
  __syncthreads();

  // ---- weight fragments from the table (v2f = two K-adjacent weights) -----
  const int m    = lane & 15;       // M (A) / N (B) index for this lane
  const int hi   = lane >> 4;       // half-wave select
  const int koff = hi * 2;          // f32 A/B layout: VGPR0 = K0|K2, VGPR1 = K1|K3

  const float* Wv = g_wtab + (ty * 16 + m) * 12;
  const float* Wh = g_wtab + (tx * 16 + m) * 12;
  v2f av[3], bh[3];
#pragma unroll
  for (int j = 0; j < 3; ++j) {
    av[j] = *(const v2f*)(Wv + 4 * j + koff);         // A: V(16x12)
    bh[j] = *(const v2f*)(Wh + 4 * j + koff);         // B: H^T(12x16)
  }

  v8f acc[NCH];
#pragma unroll
  for (int ch = 0; ch < NCH; ++ch) {
    // A = patch band, 16x12 (rows 12..15 garbage -> unused T rows)
    const float* Lc = L + ch * 192 + m * 12;
    v2f ap[3];
#pragma unroll
    for (int j = 0; j < 3; ++j)
      ap[j] = *(const v2f*)(Lc + 4 * j + koff);       // aligned ds_load_b64

    // horizontal pass: T = P_band x H^T   (three K=4 f32 WMMAs)
    v8f t = {};
#pragma unroll
    for (int j = 0; j < 3; ++j)
      t = __builtin_amdgcn_wmma_f32_16x16x4_f32(false, ap[j], false, bh[j],
                                                (short)0, t, false, false);

    // re-layout T from C/D layout (VGPR g = rows g | g+8) into B fragments
    // (VGPR0 = row 4j | 4j+2, VGPR1 = row 4j+1 | 4j+3) via ds_bpermute,
    // then vertical pass: Out = V x T
    v8f o = {};
#pragma unroll
    for (int j = 0; j < 3; ++j) {
      v2f bt;
      {
        const int rlo = 4 * j, rhi = 4 * j + 2;
        const float xlo = laneShuffle(t[rlo & 7], m + ((rlo >= 8) ? 16 : 0));
        const float xhi = laneShuffle(t[rhi & 7], m + ((rhi >= 8) ? 16 : 0));
        bt.x = hi ? xhi : xlo;
      }
      {
        const int rlo = 4 * j + 1, rhi = 4 * j + 3;
        const float ylo = laneShuffle(t[rlo & 7], m + ((rlo >= 8) ? 16 : 0));
        const float yhi = laneShuffle(t[rhi & 7], m + ((rhi >= 8) ? 16 : 0));
        bt.y = hi ? yhi : ylo;
      }
      o = __builtin_amdgcn_wmma_f32_16x16x4_f32(false, av[j], false, bt,
                                                (short)0, o, false, false);
    }
    acc[ch] = o;
  }

  // ---- store: C/D layout VGPR g holds rows g (lanes 0-15) / g+8 (16-31),
  //      col = lane&15; 3 channels -> 12 contiguous bytes per lane ----------
  const int v0 = ty * 16, o0 = tx * 16;
#pragma unroll
  for (int g = 0; g < 8; ++g) {
    const int row  = v0 + g + hi * 8;
    const int base = ((b * IMG_H + row) * IMG_W + (o0 + m)) * NCH;
    out[base + 0] = acc[0][g];
    out[base + 1] = acc[1][g];
    out[base + 2] = acc[2][g];
  }
}

extern "C" void kernel_launch(void* const* d_in, const int* in_sizes, int n_in,
                              void* d_out, int out_size, void* d_ws, size_t ws_size,
                              hipStream_t stream) {
  const float* img   = (const float*)d_in[0];
  const int*   coord = (const int*)d_in[1];
  float*       out   = (float*)d_out;
  (void)in_sizes; (void)n_in; (void)out_size; (void)d_ws; (void)ws_size;

  build_weight_table<<<2, 256, 0, stream>>>();

  const int totalTiles = NB * TPI * TPI;        // 65536 16x16 tiles
  const int blocks = totalTiles / 8;            // 8 waves (tiles) per block
  crop_resize_bicubic_wmma<<<blocks, 256, 0, stream>>>(img, coord, out);
}